// TransALoss_23304492548267
// MI455X (gfx1250) — compile-verified
//
#include <hip/hip_runtime.h>

// CDNA5 / gfx1250, wave32. fp32 WMMA (V_WMMA_F32_16X16X4_F32) keeps the
// reference's fp32 precision while exercising the matrix pipe.

typedef __attribute__((ext_vector_type(2))) float v2f;
typedef __attribute__((ext_vector_type(8))) float v8f;

static constexpr int kB = 8192;   // batch
static constexpr int kD = 128;    // embedding dim
static constexpr float kMargin = 1.0f;

// One wave (32 lanes) per batch element. 8 waves / block.
__global__ __launch_bounds__(256) void transa_scores_kernel(
    const float* __restrict__ pos_batch,   // [B,3,D]
    const float* __restrict__ neg_batch,   // [B,3,D]
    const int*   __restrict__ r_idx,       // [B]
    const float* __restrict__ rel_emb,     // [R, D*D]
    float*       __restrict__ per_batch)   // [B] relu(margin+pos-neg)
{
  __shared__ float lds[8 * 2 * kD];        // per wave: dPos[128], dNeg[128]

  const int wave = threadIdx.x >> 5;
  const int lane = threadIdx.x & 31;
  const int b    = blockIdx.x * 8 + wave;  // grid sized exactly B/8

  float* dP = &lds[wave * 2 * kD];
  float* dN = dP + kD;

  // ---- build d = |h + r - t| for positive and negative samples ----
  {
    const int j = lane * 4;                                  // 4 elems/lane
    const float4* pb = (const float4*)(pos_batch + (size_t)b * 3 * kD);
    const float4* nb = (const float4*)(neg_batch + (size_t)b * 3 * kD);
    float4 h = pb[j >> 2], r = pb[(kD >> 2) + (j >> 2)], t = pb[(kD >> 1) + (j >> 2)];
    float4 dv;
    dv.x = fabsf(h.x + r.x - t.x);
    dv.y = fabsf(h.y + r.y - t.y);
    dv.z = fabsf(h.z + r.z - t.z);
    dv.w = fabsf(h.w + r.w - t.w);
    *(float4*)(dP + j) = dv;
    h = nb[j >> 2]; r = nb[(kD >> 2) + (j >> 2)]; t = nb[(kD >> 1) + (j >> 2)];
    dv.x = fabsf(h.x + r.x - t.x);
    dv.y = fabsf(h.y + r.y - t.y);
    dv.z = fabsf(h.z + r.z - t.z);
    dv.w = fabsf(h.w + r.w - t.w);
    *(float4*)(dN + j) = dv;
  }
  __syncthreads();

  const float* W = rel_emb + (size_t)r_idx[b] * (kD * kD);
  const int half = lane >> 4;     // 0: lanes 0-15, 1: lanes 16-31
  const int col  = lane & 15;     // N index within 16-wide tile

  float sP = 0.0f, sN = 0.0f;

  // score = sum_J  (d^T W)[16J+n] * d[16J+n]
  // A (16x4) broadcasts d over rows: lane<16 -> {d[k0],d[k0+1]},
  //                                  lane>=16 -> {d[k0+2],d[k0+3]}.
  // B (4x16): v0 = W rows k0 / k0+2 (lo/hi half-lanes), v1 = rows k0+1 / k0+3.
  for (int J = 0; J < 8; ++J) {
    v8f accP = {};
    v8f accN = {};
    const float* wc = W + J * 16 + col;      // column 16J+col of W
#pragma unroll 4
    for (int k0 = 0; k0 < kD; k0 += 4) {
      const int kb = k0 + 2 * half;
      v2f bm;
      bm.x = wc[(size_t)kb * kD];            // W[kb  ][16J+col]
      bm.y = wc[(size_t)(kb + 1) * kD];      // W[kb+1][16J+col]
      v2f aP; aP.x = dP[kb]; aP.y = dP[kb + 1];
      v2f aN; aN.x = dN[kb]; aN.y = dN[kb + 1];
      // (neg_a, A, neg_b, B, c_mod, C, reuse_a, reuse_b)
      accP = __builtin_amdgcn_wmma_f32_16x16x4_f32(false, aP, false, bm,
                                                   (short)0, accP, false, false);
      accN = __builtin_amdgcn_wmma_f32_16x16x4_f32(false, aN, false, bm,
                                                   (short)0, accN, false, false);
    }
    // Every row of D equals (d^T W)[16J+n]; read row 0 (VGPR 0).
    sP += accP[0] * dP[J * 16 + col];
    sN += accN[0] * dN[J * 16 + col];
  }

  // Wave reduction; each (J,col) term is present in both half-waves -> x0.5.
  for (int off = 16; off > 0; off >>= 1) {
    sP += __shfl_xor(sP, off, 32);
    sN += __shfl_xor(sN, off, 32);
  }

  if (lane == 0) {
    const float pos = 0.5f * sP;
    const float neg = 0.5f * sN;
    per_batch[b] = fmaxf(0.0f, kMargin + pos - neg);
  }
}

// Deterministic fixed-order mean over the 8192 per-batch losses.
__global__ __launch_bounds__(256) void transa_mean_kernel(
    const float* __restrict__ per_batch, float* __restrict__ out)
{
  __shared__ float s[256];
  float acc = 0.0f;
  for (int i = threadIdx.x; i < kB; i += 256) acc += per_batch[i];
  s[threadIdx.x] = acc;
  __syncthreads();
  for (int stride = 128; stride > 0; stride >>= 1) {
    if ((int)threadIdx.x < stride) s[threadIdx.x] += s[threadIdx.x + stride];
    __syncthreads();
  }
  if (threadIdx.x == 0) out[0] = s[0] / (float)kB;
}

extern "C" void kernel_launch(void* const* d_in, const int* in_sizes, int n_in,
                              void* d_out, int out_size, void* d_ws, size_t ws_size,
                              hipStream_t stream) {
  const float* pos_batch = (const float*)d_in[0];
  const float* neg_batch = (const float*)d_in[1];
  const int*   r_idx     = (const int*)d_in[2];
  const float* rel_emb   = (const float*)d_in[3];
  float* per_batch = (float*)d_ws;          // kB floats of scratch
  float* out       = (float*)d_out;

  transa_scores_kernel<<<kB / 8, 256, 0, stream>>>(pos_batch, neg_batch, r_idx,
                                                   rel_emb, per_batch);
  transa_mean_kernel<<<1, 256, 0, stream>>>(per_batch, out);
}